// GRUD_6485400617047
// MI455X (gfx1250) — compile-verified
//
#include <hip/hip_runtime.h>

typedef __attribute__((ext_vector_type(2))) float v2f;
typedef __attribute__((ext_vector_type(8))) float v8f;

#define FDIM 21
#define TDIM 500000
#define OUTN 3

extern "C" __global__ void __launch_bounds__(32, 1)
grud_scan(const float* __restrict__ inp,
          const float* __restrict__ w_dg_x, const float* __restrict__ w_dg_h,
          const float* __restrict__ w_xz,  const float* __restrict__ w_hz,  const float* __restrict__ w_mz,
          const float* __restrict__ w_xr,  const float* __restrict__ w_hr,  const float* __restrict__ w_mr,
          const float* __restrict__ w_xh,  const float* __restrict__ w_hh,  const float* __restrict__ w_mh,
          const float* __restrict__ b_dg_x, const float* __restrict__ b_dg_h,
          const float* __restrict__ b_z,   const float* __restrict__ b_r,   const float* __restrict__ b_h,
          const float* __restrict__ w_hy,  const float* __restrict__ b_y,
          const float* __restrict__ x_mean,
          float* __restrict__ out)
{
    const int lane = threadIdx.x;                   // wave32
    const int f    = lane < FDIM ? lane : FDIM - 1; // clamp: lanes 21..31 duplicate f=20

    // Diagonal weights in registers for the whole scan.
    const float Wdgx = w_dg_x[f], Wdgh = w_dg_h[f];
    const float Bdgx = b_dg_x[f], Bdgh = b_dg_h[f];
    const float Xmean = x_mean[f];
    // Pre-scale gate weights by 0.5 (sigmoid(x)=0.5+0.5*tanh(x/2), tanh arg halved).
    const float Wxz2 = 0.5f * w_xz[f], Whz2 = 0.5f * w_hz[f], Wmz2 = 0.5f * w_mz[f], Bz2 = 0.5f * b_z[f];
    const float Wxr2 = 0.5f * w_xr[f], Whr2 = 0.5f * w_hr[f], Wmr2 = 0.5f * w_mr[f], Br2 = 0.5f * b_r[f];
    const float Wxh  = w_xh[f],  Wmh  = w_mh[f],  Bh = b_h[f];
    const float Whh2 = 0.5f * w_hh[f];   // W_hh*(r*h) == Whh2 * (tanh*h + h)

    // inp layout [3, F, T]; per-lane row pointers, 16B-aligned (T*4 % 16 == 0).
    const float4* __restrict__ Xp = (const float4*)(inp + (size_t)f * TDIM);
    const float4* __restrict__ Mp = (const float4*)(inp + (size_t)(FDIM + f) * TDIM);
    const float4* __restrict__ Dp = (const float4*)(inp + (size_t)(2 * FDIM + f) * TDIM);

    float h = 0.0f;

    // Tile = 8 timesteps = 2 float4 per array. NT = 62500 tiles.
    const int NT = TDIM / 8;
    float4 xc0 = Xp[0], xc1 = Xp[1];
    float4 mc0 = Mp[0], mc1 = Mp[1];
    float4 dc0 = Dp[0], dc1 = Dp[1];

    for (int tb = 0; tb < NT; ++tb) {
        // Unconditional next-tile load for all but the last iteration is arranged
        // by loading (tb+1) clamped to NT-1: branch-free, always in-bounds.
        const int nb = (tb + 1 < NT) ? (tb + 1) : (NT - 1);
        float4 xn0 = Xp[2 * nb],     mn0 = Mp[2 * nb],     dn0 = Dp[2 * nb];
        float4 xn1 = Xp[2 * nb + 1], mn1 = Mp[2 * nb + 1], dn1 = Dp[2 * nb + 1];

        // Deep speculative prefetch (~8 tiles = 64 steps ahead); OOB prefetches
        // are silently dropped, so no guard branch needed.
        __builtin_prefetch(&Xp[2 * tb + 16], 0, 1);   // global_prefetch_b8
        __builtin_prefetch(&Mp[2 * tb + 16], 0, 1);
        __builtin_prefetch(&Dp[2 * tb + 16], 0, 1);

        const float xs[8] = {xc0.x, xc0.y, xc0.z, xc0.w, xc1.x, xc1.y, xc1.z, xc1.w};
        const float ms[8] = {mc0.x, mc0.y, mc0.z, mc0.w, mc1.x, mc1.y, mc1.z, mc1.w};
        const float ds[8] = {dc0.x, dc0.y, dc0.z, dc0.w, dc1.x, dc1.y, dc1.z, dc1.w};

        #pragma unroll
        for (int k = 0; k < 8; ++k) {
            const float x = xs[k], m = ms[k], d = ds[k];
            // ---- off-critical-path (input-only) work ----
            const float gx = __expf(-fmaxf(fmaf(Wdgx, d, Bdgx), 0.0f));  // v_exp_f32
            const float gh = __expf(-fmaxf(fmaf(Wdgh, d, Bdgh), 0.0f));
            const float p  = fmaf(gx, x - Xmean, Xmean);   // gx*x + (1-gx)*x_mean
            const float xi = fmaf(m, x - p, p);            // m*x + (1-m)*p
            const float az2 = fmaf(Wxz2, xi, fmaf(Wmz2, m, Bz2));
            const float ar2 = fmaf(Wxr2, xi, fmaf(Wmr2, m, Br2));
            const float ah  = fmaf(Wxh,  xi, fmaf(Wmh,  m, Bh));
            // ---- serial h-chain: 6 VALU + 2 v_tanh_f32 deep ----
            h = gh * h;
            const float tz = __builtin_amdgcn_tanhf(fmaf(Whz2, h, az2));
            const float z  = fmaf(0.5f, tz, 0.5f);                     // sigmoid
            const float tr = __builtin_amdgcn_tanhf(fmaf(Whr2, h, ar2));
            const float u  = fmaf(tr, h, h);                           // 2*r*h
            const float ht = __builtin_amdgcn_tanhf(fmaf(Whh2, u, ah));
            h = fmaf(z, ht - h, h);                                    // (1-z)h + z*ht
        }
        xc0 = xn0; xc1 = xn1; mc0 = mn0; mc1 = mn1; dc0 = dn0; dc1 = dn1;
    }

    // ---- out = w_hy(3x21) @ h(21) + b_y via V_WMMA_F32_16X16X4_F32 ----
    // A (16x4 f32): lanes 0-15 hold row M=l, {K0,K1}; lanes 16-31 row M=l-16, {K2,K3}.
    // B (4x16 f32): mirrored K split, N = lane%16 (h broadcast across columns).
    // D (16x16 f32): VGPR v, lane l -> M = v + (l>=16 ? 8 : 0), N = l%16.
    const int half   = lane >> 4;
    const int row    = lane & 15;
    const int rclamp = row < OUTN ? row : 0;

    v8f acc = {};
    #pragma unroll
    for (int c = 0; c < 6; ++c) {          // K = 24 >= 21, zero-padded
        const int k0  = 4 * c + 2 * half;
        const int k1  = k0 + 1;
        const int k0c = k0 < FDIM ? k0 : 0;
        const int k1c = k1 < FDIM ? k1 : 0;
        // Branchless: clamped unconditional loads + v_cndmask selects.
        const float w0 = w_hy[rclamp * FDIM + k0c];
        const float w1 = w_hy[rclamp * FDIM + k1c];
        const bool  a0v = (row < OUTN) & (k0 < FDIM);
        const bool  a1v = (row < OUTN) & (k1 < FDIM);
        const float h0 = __shfl(h, k0c);   // ds_bpermute_b32
        const float h1 = __shfl(h, k1c);
        v2f a, b;
        a.x = a0v ? w0 : 0.0f;
        a.y = a1v ? w1 : 0.0f;
        b.x = (k0 < FDIM) ? h0 : 0.0f;
        b.y = (k1 < FDIM) ? h1 : 0.0f;
        acc = __builtin_amdgcn_wmma_f32_16x16x4_f32(
            /*neg_a=*/false, a, /*neg_b=*/false, b,
            /*c_mod=*/(short)0, acc, /*reuse_a=*/false, /*reuse_b=*/false);
    }

    if (lane == 0) {                        // D[m][0] -> lane 0, VGPRs 0..2
        out[0] = acc[0] + b_y[0];
        out[1] = acc[1] + b_y[1];
        out[2] = acc[2] + b_y[2];
    }
}

extern "C" void kernel_launch(void* const* d_in, const int* in_sizes, int n_in,
                              void* d_out, int out_size, void* d_ws, size_t ws_size,
                              hipStream_t stream) {
    (void)in_sizes; (void)n_in; (void)d_ws; (void)ws_size; (void)out_size;
    const float* inp = (const float*)d_in[0];
    const float* p[19];
    for (int i = 0; i < 19; ++i) p[i] = (const float*)d_in[1 + i];
    // p[0..15] = names_F in dict order, p[16] = w_hy, p[17] = b_y, p[18] = x_mean
    grud_scan<<<1, 32, 0, stream>>>(
        inp,
        p[0],  p[1],  p[2],  p[3],  p[4],
        p[5],  p[6],  p[7],  p[8],  p[9],
        p[10], p[11], p[12], p[13], p[14], p[15],
        p[16], p[17], p[18],
        (float*)d_out);
}